// Attn_5093831213229
// MI455X (gfx1250) — compile-verified
//
#include <hip/hip_runtime.h>
#include <math.h>

// Problem constants (match reference)
#define H 512
#define SEQ 4096
#define BATCH 32
#define NROWS (SEQ * BATCH) /* 131072 flat (s,b) rows */

typedef __attribute__((ext_vector_type(2))) float v2f;
typedef __attribute__((ext_vector_type(8))) float v8f;

// ---------------------------------------------------------------------------
// Kernel 1: u[h] = sum_k v[k] * W[k, H+h]   (W is row-major H x 2H)
// Consecutive h -> consecutive addresses: perfectly coalesced column read.
// ---------------------------------------------------------------------------
__global__ __launch_bounds__(256) void attn_u_kernel(const float* __restrict__ W,
                                                     const float* __restrict__ v,
                                                     float* __restrict__ u) {
  int h = blockIdx.x * blockDim.x + threadIdx.x;
  if (h >= H) return;
  const float* wp = W + H + h; // column H+h, stride 2H floats
  float acc = 0.0f;
#pragma unroll 8
  for (int k = 0; k < H; ++k) acc = fmaf(v[k], wp[(size_t)k * (2 * H)], acc);
  u[h] = acc;
}

// ---------------------------------------------------------------------------
// Kernel 2: scores_t[b*SEQ + s] = sum_h E[s,b,h] * u[h]
// One wave32 handles 16 consecutive flat rows (flat = s*32 + b).
//
// WMMA role swap vs naive: A = u broadcast into all 16 M-rows, B = encoder
// data (4x16). Per CDNA5 ISA layouts:
//   A (16x4 f32): lanes 0-15 hold {K0,K1} (any M), lanes 16-31 hold {K2,K3}
//     -> u values, identical for all M rows: trivial per-lane pattern from LDS.
//   B (4x16 f32): lane n holds {K0,K1} of column n; lane n+16 holds {K2,K3}
//     -> each lane does ONE global_load_b64 at row_n + h0 + (hi?2:0).
//        No cross-lane exchange (no ds_bpermute), no cndmask reassembly.
//   D (16x16 f32): VGPR0, lanes 0-15 = D[M=0, N=lane]. All M rows identical,
//     so lane n (n<16) holds score of tile-row n in c[0]. Clean epilogue.
//
// C-operand chaining accumulates K=512 over 128 WMMAs. Memory-bound: 256 MB
// streamed at HBM rate; WMMA columns are free capacity.
// ---------------------------------------------------------------------------
__global__ __launch_bounds__(256) void attn_scores_kernel(const float* __restrict__ E,
                                                          const float* __restrict__ u,
                                                          float* __restrict__ scores_t) {
  __shared__ float lds_u[H];
  const int t = threadIdx.x;
  lds_u[t] = u[t];
  lds_u[t + 256] = u[t + 256];
  __syncthreads();

  const int lane = t & 31;
  const bool hi = lane >= 16;                              // K=2,3 half
  const long wave = (long)((blockIdx.x * 256 + t) >> 5);   // 0..8191
  const long base = wave * 16;                             // first flat row of tile
  const int n = lane & 15;                                 // tile-row / B column
  // Per-lane stream pointer: row n of the tile, offset into the K-half.
  const float* lanep = E + (base + n) * (long)H + (hi ? 2 : 0);

  v8f c = {}; // f32 accumulator (16x16, all M rows identical)

#pragma unroll 4
  for (int i = 0; i < H / 8; ++i) {
    const int h0 = i * 8;
    // B operands: direct b64 loads, no shuffle needed.
    const float2 e1 = *(const float2*)(lanep + h0);       // K = h0..h0+3 half
    const float2 e2 = *(const float2*)(lanep + h0 + 4);   // K = h0+4..h0+7 half
    __builtin_prefetch(lanep + h0 + 64, 0, 1);            // 256B ahead (global_prefetch_b8)

    // A operands: u broadcast (same for every M row), from LDS.
    const int ua = h0 + (hi ? 2 : 0);
    v2f a1, a2, b1, b2;
    a1.x = lds_u[ua];      a1.y = lds_u[ua + 1];
    a2.x = lds_u[ua + 4];  a2.y = lds_u[ua + 5];
    b1.x = e1.x;           b1.y = e1.y;
    b2.x = e2.x;           b2.y = e2.y;

    c = __builtin_amdgcn_wmma_f32_16x16x4_f32(false, a1, false, b1, (short)0, c, false, false);
    c = __builtin_amdgcn_wmma_f32_16x16x4_f32(false, a2, false, b2, (short)0, c, false, false);
  }

  // Lane n (n < 16) holds score of flat row (base + n) in c[0].
  if (!hi) {
    const long r = base + n;            // flat row = s*32 + b
    const int b = (int)(r & (BATCH - 1));
    const long s = r >> 5;
    scores_t[(long)b * SEQ + s] = c[0];
  }
}

// ---------------------------------------------------------------------------
// Kernel 3: out[b, 0, s] = softmax_s(scores_t[b, s]); one block per batch b.
// ---------------------------------------------------------------------------
__global__ __launch_bounds__(256) void attn_softmax_kernel(const float* __restrict__ scores_t,
                                                           float* __restrict__ out) {
  __shared__ float red[256];
  const int b = blockIdx.x;
  const int t = threadIdx.x;
  const float* row = scores_t + (long)b * SEQ;
  float* orow = out + (long)b * SEQ;

  // Pass 1: max
  float m = -3.4e38f;
  for (int s = t; s < SEQ; s += 256) m = fmaxf(m, row[s]);
  red[t] = m;
  __syncthreads();
  for (int w = 128; w > 0; w >>= 1) {
    if (t < w) red[t] = fmaxf(red[t], red[t + w]);
    __syncthreads();
  }
  m = red[0];
  __syncthreads();

  // Pass 2: exp and sum
  float sum = 0.0f;
  for (int s = t; s < SEQ; s += 256) {
    float e = __expf(row[s] - m);
    orow[s] = e;
    sum += e;
  }
  red[t] = sum;
  __syncthreads();
  for (int w = 128; w > 0; w >>= 1) {
    if (t < w) red[t] += red[t + w];
    __syncthreads();
  }
  const float inv = 1.0f / red[0];

  // Pass 3: normalize
  for (int s = t; s < SEQ; s += 256) orow[s] *= inv;
}

// ---------------------------------------------------------------------------
// Launch. Inputs: [hidden, encoderOutputs, W, b, v]. hidden and b are
// constant per softmax row and cancel exactly; never read. Workspace:
//   ws[0 .. 511]            : u (f32)
//   ws[512 .. 512+131071]   : scores transposed [b][s] (f32)
// ---------------------------------------------------------------------------
extern "C" void kernel_launch(void* const* d_in, const int* in_sizes, int n_in,
                              void* d_out, int out_size, void* d_ws, size_t ws_size,
                              hipStream_t stream) {
  (void)in_sizes; (void)n_in; (void)out_size; (void)ws_size;
  const float* E = (const float*)d_in[1]; // encoderOutputs (4096,32,512)
  const float* W = (const float*)d_in[2]; // (512,1024)
  const float* v = (const float*)d_in[4]; // (512,)
  float* out = (float*)d_out;             // (32,1,4096) flat

  float* u = (float*)d_ws;
  float* scores_t = u + H;

  attn_u_kernel<<<2, 256, 0, stream>>>(W, v, u);
  // 131072 rows / 16 rows-per-wave / 8 waves-per-block = 1024 blocks
  attn_scores_kernel<<<(NROWS / 16) / 8, 256, 0, stream>>>(E, u, scores_t);
  attn_softmax_kernel<<<BATCH, 256, 0, stream>>>(scores_t, out);
}